// Attention_39256001085540
// MI455X (gfx1250) — compile-verified
//
#include <hip/hip_runtime.h>
#include <hip/hip_bf16.h>
#include <math.h>

// ---------------------------------------------------------------------------
// Tacotron2-style location-sensitive attention, MI455X (gfx1250, wave32).
// B=64, T=2048, E=512, D_LSTM=1024, D_ATT=128, C_LOC=32, K=31.
//
// Roofline: ~335 MB streamed (enc 268 MB + peo 67 MB) => ~14 us @ 23.3 TB/s;
// ~1.5 GFLOP fp32 => memory-bound. GEMM pieces run on V_WMMA_F32_16X16X4_F32
// (full fp32 precision); the two big streams use non-temporal loads (enc 64-bit, peo 32-bit) so
// they bypass/minimize L2 residency (L2 = 192 MB, reused operands are tiny).
// The dominant context bmm is split 16-way over T (1024 WGs) to keep enough
// loads in flight to saturate HBM, with a small partial-sum reduce at the end.
// ---------------------------------------------------------------------------

#define B_      64
#define T_      2048
#define E_      512
#define DL_     1024
#define DA_     128
#define CL_     32
#define KW_     31
#define TSPLIT_ 16

typedef float v2f __attribute__((ext_vector_type(2)));
typedef float v8f __attribute__((ext_vector_type(8)));

__device__ __forceinline__ v8f wmma_f32_16x16x4(v2f a, v2f b, v8f c) {
    // 8 args: (neg_a, A, neg_b, B, c_mod, C, reuse_a, reuse_b)
    return __builtin_amdgcn_wmma_f32_16x16x4_f32(
        false, a, false, b, (short)0, c, false, false);
}

// ---------------------------------------------------------------------------
// Kernel 1: pl[b,a] = sum_d lstm[b,d] * W_lstm[a,d]
// M=B_=64 (b), N=DA_=128 (a), K=DL_=1024. Grid (4,8), block = 1 wave32.
// A lane layout (16x4 f32): lane l -> row m=l&15, K = (l>>4)*2 + j (j=0,1)
// B lane layout (4x16 f32): lane l -> col n=l&15, K = (l>>4)*2 + j
// D layout: lane l -> col n0+(l&15), rows m0 + (l>>4)*8 + r in slot r.
// ---------------------------------------------------------------------------
__global__ void __launch_bounds__(32)
pl_gemm_kernel(const float* __restrict__ lstm,   // [B, DL]
               const float* __restrict__ Wl,     // [DA, DL]
               float* __restrict__ pl)           // [B, DA]
{
    const int lane = threadIdx.x;
    const int half = lane >> 4;
    const int l    = lane & 15;
    const int m0   = blockIdx.x * 16;   // over B_
    const int n0   = blockIdx.y * 16;   // over DA_

    const float* arow = lstm + (size_t)(m0 + l) * DL_;
    const float* brow = Wl   + (size_t)(n0 + l) * DL_;

    v8f acc = {};
    for (int k0 = 0; k0 < DL_; k0 += 4) {
        const int ka = k0 + half * 2;
        v2f a = { arow[ka], arow[ka + 1] };
        v2f b = { brow[ka], brow[ka + 1] };
        acc = wmma_f32_16x16x4(a, b, acc);
    }
#pragma unroll
    for (int r = 0; r < 8; ++r)
        pl[(size_t)(m0 + half * 8 + r) * DA_ + (n0 + l)] = acc[r];
}

// ---------------------------------------------------------------------------
// Kernel 2: fused location-conv + location-dense GEMM + tanh + energy dot.
// One wave32 per (b, 16-row t-tile). Grid (T/16, B).
//   conv:   Atile[m][c] = conv_b[c] + sum_j awc[b, t0+m+j-15] * conv_w[c,j]
//   gemm:   dense[m,a]  = sum_c Atile[m][c] * W_dense[a,c]      (64 WMMAs)
//   energy: e[t] = b_e + sum_a w_e[a]*tanh(pl[b,a]+dense+b_d[a]+peo[b,t,a])
//   mask:   t >= text_lengths[b] -> -inf
// peo is streamed once -> non-temporal loads.
// ---------------------------------------------------------------------------
__global__ void __launch_bounds__(32)
energy_kernel(const float* __restrict__ awc,    // [B, T]
              const float* __restrict__ cw,     // [CL, 1, KW]
              const float* __restrict__ cb,     // [CL]
              const float* __restrict__ Wd,     // [DA, CL]
              const float* __restrict__ bd,     // [DA]
              const float* __restrict__ pl,     // [B, DA]
              const float* __restrict__ peo,    // [B, T, DA]
              const float* __restrict__ we,     // [1, DA]
              const float* __restrict__ be,     // [1]
              const int*   __restrict__ tlen,   // [B]
              float* __restrict__ energies)     // [B, T]
{
    const int lane = threadIdx.x;
    const int half = lane >> 4;
    const int l    = lane & 15;
    const int b    = blockIdx.y;
    const int t0   = blockIdx.x * 16;

    __shared__ float awc_s[16 + KW_ - 1];       // 46 taps
    __shared__ float Atile[16][CL_ + 1];        // +1 pad vs bank conflicts

    // Stage the 46 cum-attention taps this tile needs (zero-padded 'same').
    for (int i = lane; i < 16 + KW_ - 1; i += 32) {
        const int tt = t0 - (KW_ / 2) + i;
        awc_s[i] = (tt >= 0 && tt < T_) ? awc[(size_t)b * T_ + tt] : 0.0f;
    }
    __syncthreads();

    // 16x32 conv tile: 512 outputs, 16 per lane, 31 MACs each.
    for (int i = lane; i < 16 * CL_; i += 32) {
        const int m = i >> 5;
        const int c = i & (CL_ - 1);
        float s = cb[c];
#pragma unroll
        for (int j = 0; j < KW_; ++j)
            s += awc_s[m + j] * cw[c * KW_ + j];
        Atile[m][c] = s;
    }
    __syncthreads();

    // [16 x 32] * [32 x 128] on fp32 WMMA: 8 N-tiles x 8 K-steps.
    v8f acc[8];
#pragma unroll
    for (int n = 0; n < 8; ++n) acc[n] = (v8f){};

#pragma unroll
    for (int k0 = 0; k0 < CL_; k0 += 4) {
        const int ka = k0 + half * 2;
        v2f a = { Atile[l][ka], Atile[l][ka + 1] };
#pragma unroll
        for (int n = 0; n < 8; ++n) {
            const float* wrow = Wd + (size_t)(n * 16 + l) * CL_;
            v2f bf = { wrow[ka], wrow[ka + 1] };
            acc[n] = wmma_f32_16x16x4(a, bf, acc[n]);
        }
    }

    // Epilogue: lane holds column a = n*16 + l, rows t0 + half*8 + r.
    float erow[8];
#pragma unroll
    for (int r = 0; r < 8; ++r) erow[r] = 0.0f;

#pragma unroll
    for (int n = 0; n < 8; ++n) {
        const int a_col = n * 16 + l;
        const float plv = pl[(size_t)b * DA_ + a_col];
        const float bdv = bd[a_col];
        const float wev = we[a_col];
#pragma unroll
        for (int r = 0; r < 8; ++r) {
            const int t = t0 + half * 8 + r;
            const float pv = __builtin_nontemporal_load(
                &peo[((size_t)b * T_ + t) * DA_ + a_col]);
            const float v = acc[n][r] + plv + bdv + pv;
            erow[r] += wev * tanhf(v);
        }
    }

    // Reduce across the 16 lanes of each half (xor 1,2,4,8 stays in-half).
#pragma unroll
    for (int off = 1; off < 16; off <<= 1) {
#pragma unroll
        for (int r = 0; r < 8; ++r)
            erow[r] += __shfl_xor(erow[r], off, 32);
    }

    if (l == 0) {
        const int   len = tlen[b];
        const float beo = be[0];
#pragma unroll
        for (int r = 0; r < 8; ++r) {
            const int t = t0 + half * 8 + r;
            const float e = erow[r] + beo;
            energies[(size_t)b * T_ + t] =
                (t < len) ? e : -__builtin_inff();
        }
    }
}

// ---------------------------------------------------------------------------
// Kernel 3: softmax over T per batch row. One block per b.
// ---------------------------------------------------------------------------
__global__ void __launch_bounds__(256)
softmax_kernel(const float* __restrict__ energies,  // [B, T]
               float* __restrict__ weights)         // [B, T]
{
    __shared__ float red[256];
    const int b   = blockIdx.x;
    const int tid = threadIdx.x;
    const float* e = energies + (size_t)b * T_;

    float m = -__builtin_inff();
    for (int t = tid; t < T_; t += 256) m = fmaxf(m, e[t]);
    red[tid] = m;
    __syncthreads();
    for (int s = 128; s > 0; s >>= 1) {
        if (tid < s) red[tid] = fmaxf(red[tid], red[tid + s]);
        __syncthreads();
    }
    m = red[0];
    __syncthreads();

    float sum = 0.0f;
    for (int t = tid; t < T_; t += 256) sum += expf(e[t] - m);
    red[tid] = sum;
    __syncthreads();
    for (int s = 128; s > 0; s >>= 1) {
        if (tid < s) red[tid] += red[tid + s];
        __syncthreads();
    }
    const float inv = 1.0f / red[0];

    for (int t = tid; t < T_; t += 256)
        weights[(size_t)b * T_ + t] = expf(e[t] - m) * inv;
}

// ---------------------------------------------------------------------------
// Kernel 4a: split-T partial context.
//   part[s][b*E + e] = sum_{t in split s} w[b,t] * enc[b,t,e]
// Grid (1, B, TSPLIT) = 1024 WGs x 256 threads; each thread owns one float2
// column pair -> 64-bit coalesced non-temporal loads of the 268 MB stream.
// ---------------------------------------------------------------------------
__global__ void __launch_bounds__(256)
context_partial_kernel(const float* __restrict__ w,    // [B, T]
                       const float* __restrict__ enc,  // [B, T, E]
                       float* __restrict__ part)       // [TSPLIT, B*E]
{
    const int b  = blockIdx.y;
    const int s  = blockIdx.z;
    const int e2 = threadIdx.x;                       // float2 column, 0..E/2-1
    const int tb = s * (T_ / TSPLIT_);                // 128 rows per split

    const v2f* er = reinterpret_cast<const v2f*>(enc + (size_t)b * T_ * E_) + e2;
    const float* wr = w + (size_t)b * T_ + tb;

    v2f acc = { 0.0f, 0.0f };
    for (int t = 0; t < T_ / TSPLIT_; ++t) {
        if ((t & 31) == 0 && t + 32 < T_ / TSPLIT_)
            __builtin_prefetch(&er[(size_t)(tb + t + 32) * (E_ / 2)], 0, 0);
        const float wt = wr[t];
        const v2f   v  = __builtin_nontemporal_load(&er[(size_t)(tb + t) * (E_ / 2)]);
        acc[0] += wt * v[0];
        acc[1] += wt * v[1];
    }

    v2f* p = reinterpret_cast<v2f*>(part + (size_t)s * B_ * E_)
           + (size_t)b * (E_ / 2) + e2;
    *p = acc;
}

// ---------------------------------------------------------------------------
// Kernel 4b: reduce the TSPLIT partials -> attention_context [B, E].
// ---------------------------------------------------------------------------
__global__ void __launch_bounds__(256)
context_reduce_kernel(const float* __restrict__ part,  // [TSPLIT, B*E]
                      float* __restrict__ ctx)         // [B*E]
{
    const int i = blockIdx.x * 256 + threadIdx.x;      // 0 .. B*E-1
    float acc = 0.0f;
#pragma unroll
    for (int s = 0; s < TSPLIT_; ++s)
        acc += part[(size_t)s * B_ * E_ + i];
    ctx[i] = acc;
}

// ---------------------------------------------------------------------------
extern "C" void kernel_launch(void* const* d_in, const int* in_sizes, int n_in,
                              void* d_out, int out_size, void* d_ws, size_t ws_size,
                              hipStream_t stream) {
    const float* enc  = (const float*)d_in[0];   // [B,T,E]
    const int*   tlen = (const int*)  d_in[1];   // [B]
    const float* peo  = (const float*)d_in[2];   // [B,T,DA]
    const float* lstm = (const float*)d_in[3];   // [B,1,DL]
    const float* awc  = (const float*)d_in[4];   // [B,T]
    const float* Wl   = (const float*)d_in[5];   // [DA,DL]
    const float* cw   = (const float*)d_in[6];   // [CL,1,KW]
    const float* cb   = (const float*)d_in[7];   // [CL]
    const float* Wd   = (const float*)d_in[8];   // [DA,CL]
    const float* bd   = (const float*)d_in[9];   // [DA]
    const float* we   = (const float*)d_in[10];  // [1,DA]
    const float* be   = (const float*)d_in[11];  // [1]

    float* ctx_out = (float*)d_out;              // [B,E]   (output 0)
    float* w_out   = (float*)d_out + B_ * E_;    // [B,T]   (output 1)

    float* pl_ws   = (float*)d_ws;               // [B,DA]        32 KB
    float* e_ws    = pl_ws + B_ * DA_;           // [B,T]         512 KB
    float* part_ws = e_ws + B_ * T_;             // [TSPLIT,B*E]  2 MB

    pl_gemm_kernel<<<dim3(B_ / 16, DA_ / 16), 32, 0, stream>>>(lstm, Wl, pl_ws);
    energy_kernel<<<dim3(T_ / 16, B_), 32, 0, stream>>>(
        awc, cw, cb, Wd, bd, pl_ws, peo, we, be, tlen, e_ws);
    softmax_kernel<<<B_, 256, 0, stream>>>(e_ws, w_out);
    context_partial_kernel<<<dim3(1, B_, TSPLIT_), 256, 0, stream>>>(
        w_out, enc, part_ws);
    context_reduce_kernel<<<(B_ * E_) / 256, 256, 0, stream>>>(part_ws, ctx_out);
}